// DistMultDecoder_90091234000907
// MI455X (gfx1250) — compile-verified
//
#include <hip/hip_runtime.h>

typedef __attribute__((ext_vector_type(2))) float v2f;
typedef __attribute__((ext_vector_type(8))) float v8f;

// One wave32 handles 16 edges. Score vector = diagonal of a 16x16 f32 WMMA:
//   A[m][k] = xl[m][k] * rel[m][k]   (16x4 per chunk, 8 chunks over D=32)
//   B[k][n] = xr[n][k]
//   C = A x B  (accumulated over chunks);  score[e] = C[e][e]
__global__ __launch_bounds__(256) void distmult_wmma_kernel(
    const float* __restrict__ x,
    const float* __restrict__ R,
    const int*   __restrict__ eidx,    // [2, E] row-major
    const int*   __restrict__ etype,   // [E]
    float*       __restrict__ out,     // [E]
    int n_edges)
{
    const int lane        = threadIdx.x & 31;
    const int waveInBlock = threadIdx.x >> 5;
    const long long group = (long long)blockIdx.x * (blockDim.x >> 5) + waveInBlock;
    const long long ebase = group * 16;
    if (ebase >= (long long)n_edges) return;   // wave-uniform exit; EXEC stays all-1s below

    // ISA 32-bit A 16x4 layout: lane L -> row m = L&15, K-pair selected by h = L>>4.
    const int m = lane & 15;
    const int h = lane >> 4;

    long long e = ebase + m;
    if (e >= (long long)n_edges) e = (long long)n_edges - 1;  // tail clamp (loads only)

    const int li = eidx[e];
    const int ri = eidx[(long long)n_edges + e];
    const int rt = etype[e];

    const float* __restrict__ xlrow = x + (long long)li * 32;  // 128B row == 1 cacheline
    const float* __restrict__ xrrow = x + (long long)ri * 32;
    const float* __restrict__ rrow  = R + (long long)rt * 32;  // L0/L2-hot (sorted etype)

    v8f acc = {0.f, 0.f, 0.f, 0.f, 0.f, 0.f, 0.f, 0.f};

    #pragma unroll
    for (int c = 0; c < 8; ++c) {
        const int k = c * 4 + h * 2;                 // lanes 0-15: K=4c..4c+1; 16-31: 4c+2..4c+3
        v2f xl2 = *(const v2f*)(xlrow + k);          // global_load_b64, 8B aligned
        v2f rl2 = *(const v2f*)(rrow  + k);
        v2f xr2 = *(const v2f*)(xrrow + k);
        v2f a;
        a.x = xl2.x * rl2.x;                         // fold diagonal relation into A
        a.y = xl2.y * rl2.y;
        // (neg_a, A, neg_b, B, c_mod, C, reuse_a, reuse_b) -> v_wmma_f32_16x16x4_f32
        acc = __builtin_amdgcn_wmma_f32_16x16x4_f32(
            false, a, false, xr2, (short)0, acc, false, false);
    }

    // Diagonal extraction per documented C/D layout:
    //   C[v][v]     lives at (vgpr v, lane v),      v = 0..7
    //   C[v+8][v+8] lives at (vgpr v, lane v+24)
    float s = 0.0f;
    #pragma unroll
    for (int v = 0; v < 8; ++v) {
        if (lane == v)      s = acc[v];
        if (lane == v + 24) s = acc[v];
    }

    const bool holds   = (lane < 8) || (lane >= 24);
    const long long eo = ebase + (long long)((lane < 8) ? lane : (lane - 16));
    if (holds && eo < (long long)n_edges) {
        out[eo] = 1.0f / (1.0f + __expf(-s));        // sigmoid
    }
}

extern "C" void kernel_launch(void* const* d_in, const int* in_sizes, int n_in,
                              void* d_out, int out_size, void* d_ws, size_t ws_size,
                              hipStream_t stream) {
    (void)n_in; (void)out_size; (void)d_ws; (void)ws_size;

    const float* x     = (const float*)d_in[0];   // [100000, 32]
    const float* R     = (const float*)d_in[1];   // [964, 32]
    const int*   eidx  = (const int*)  d_in[2];   // [2, E]
    const int*   etype = (const int*)  d_in[3];   // [E]
    float*       out   = (float*)d_out;           // [E]

    const int E = in_sizes[3];                    // edge_type element count == E
    const int groups        = (E + 15) / 16;      // 16 edges per wave
    const int wavesPerBlock = 8;                  // 256 threads = 8 wave32
    const int blocks        = (groups + wavesPerBlock - 1) / wavesPerBlock;

    distmult_wmma_kernel<<<blocks, 256, 0, stream>>>(x, R, eidx, etype, out, E);
}